// GATNSR_74947179315826
// MI455X (gfx1250) — compile-verified
//
#include <hip/hip_runtime.h>

// ---------------------------------------------------------------------------
// MI455X / gfx1250 forward pass for GAT-based social recommendation.
// Strategy: factor GATs into node-level WMMA GEMMs + scalar edge phases.
//   - node transforms H = X@W + b  -> v_wmma_f32_16x16x4_f32 tiles
//   - implicit concat GEMMs done as two accumulate passes (no split branch)
//   - attention logits reduced to per-node scalars (no per-edge matmul)
//   - segment max/sum via ordered-uint atomicMax + global_atomic_add_f32
// ---------------------------------------------------------------------------

typedef __attribute__((ext_vector_type(2))) float v2f;
typedef __attribute__((ext_vector_type(8))) float v8f;

#define WAVES_PER_BLOCK 8
#define DM 64  // embedding dim

// ---- ordered-uint encoding for float atomic max -----------------------------
__device__ __forceinline__ unsigned encOrd(float f) {
  unsigned u = __float_as_uint(f);
  return (u >> 31) ? ~u : (u | 0x80000000u);
}
__device__ __forceinline__ float decOrd(unsigned u) {
  u = (u >> 31) ? (u & 0x7FFFFFFFu) : ~u;
  return __uint_as_float(u);
}

// ---- tiled WMMA f32 GEMM: C = act((ACC ? C : 0) + A @ W + bias) -------------
// A row-major [M,K], W row-major [K,N], bias [N]. One 16x16 tile per wave.
// Fragment layouts per CDNA5 ISA (f32, 16x16x4):
//   A 16x4 : lane l<16 -> M=l,   K=k0..k0+1 ; lane>=16 -> M=l-16, K=k0+2..k0+3
//   B 4x16 : VGPR j, half h     -> K=k0+2h+j, N=l16
//   C/D    : VGPR v, half h     -> M=v+8h,    N=l16
template <int K, int RELU, int ACC>
__global__ void gemm_wmma_f32(const float* __restrict__ A, const float* __restrict__ W,
                              const float* __restrict__ bias, float* __restrict__ C,
                              int M, int N) {
  const int lane = threadIdx.x & 31;
  const int wave = threadIdx.x >> 5;
  const int nTiles = N >> 4;
  const int tiles = (M >> 4) * nTiles;
  const int tile = blockIdx.x * WAVES_PER_BLOCK + wave;
  if (tile >= tiles) return;                  // wave-uniform: EXEC stays all-1s
  const int tm = tile / nTiles;
  const int tn = tile - tm * nTiles;
  const int half = lane >> 4;
  const int l16 = lane & 15;
  const int rowA = tm * 16 + l16;
  const int colB = tn * 16 + l16;
  const float* aRow = A + (size_t)rowA * K;   // per-lane contiguous K stream

  v8f acc = {};
  if (ACC) {
#pragma unroll
    for (int v = 0; v < 8; ++v)
      acc[v] = C[(size_t)(tm * 16 + v + (half << 3)) * N + colB];
  }
#pragma unroll 8
  for (int k0 = 0; k0 < K; k0 += 4) {
    const int ka = k0 + (half << 1);          // lanes 16-31 take K=k0+2,k0+3
    const v2f a = *reinterpret_cast<const v2f*>(aRow + ka);   // global_load_b64
    v2f b;
    b.x = W[(size_t)ka * N + colB];
    b.y = W[(size_t)(ka + 1) * N + colB];
    acc = __builtin_amdgcn_wmma_f32_16x16x4_f32(false, a, false, b,
                                                (short)0, acc, false, false);
  }
  const float bv = bias ? bias[colB] : 0.0f;
#pragma unroll
  for (int v = 0; v < 8; ++v) {
    float c = acc[v] + bv;
    if (RELU) c = fmaxf(c, 0.0f);
    C[(size_t)(tm * 16 + v + (half << 3)) * N + colB] = c;
  }
}

template <int K, int RELU, int ACC>
static void launch_gemm(const float* A, const float* W, const float* bias, float* C,
                        int M, int N, hipStream_t stream) {
  const int tiles = (M / 16) * (N / 16);
  const int blocks = (tiles + WAVES_PER_BLOCK - 1) / WAVES_PER_BLOCK;
  gemm_wmma_f32<K, RELU, ACC>
      <<<blocks, 32 * WAVES_PER_BLOCK, 0, stream>>>(A, W, bias, C, M, N);
}

// ---- per-node attention scores: sT = H.a_w[0:64], sS = H.a_w[64:128] --------
__global__ void node_scores(const float* __restrict__ H, const float* __restrict__ a_w,
                            int n, float* __restrict__ sT, float* __restrict__ sS) {
  const int gid = blockIdx.x * blockDim.x + threadIdx.x;
  const int wid = gid >> 5;
  const int lane = gid & 31;
  if (wid >= n) return;
  const float* h = H + (size_t)wid * DM;
  const float h0 = h[lane], h1 = h[lane + 32];
  float t = h0 * a_w[lane] + h1 * a_w[lane + 32];
  float s = h0 * a_w[64 + lane] + h1 * a_w[96 + lane];
  for (int off = 16; off; off >>= 1) {
    t += __shfl_down(t, off);
    s += __shfl_down(s, off);
  }
  if (lane == 0) { sT[wid] = t; sS[wid] = s; }
}

// ---- rating embedding transforms (8x64 each, tiny) --------------------------
__global__ void rating_hr(const float* __restrict__ remb, const float* __restrict__ W,
                          float* __restrict__ HR) {
  const int tid = blockIdx.x * blockDim.x + threadIdx.x;
  if (tid >= 8 * DM) return;
  const int r = tid >> 6, d = tid & 63;
  float acc = 0.0f;
  for (int k = 0; k < DM; ++k) acc += remb[r * DM + k] * W[k * DM + d];
  HR[tid] = acc;   // no bias: (s+r)@W + b = (s@W+b) + r@W
}
__global__ void rating_score(const float* __restrict__ HR, const float* __restrict__ a_w,
                             float* __restrict__ SrS) {
  const int r = threadIdx.x;
  if (r >= 8) return;
  float acc = 0.0f;
  for (int d = 0; d < DM; ++d) acc += HR[r * DM + d] * a_w[64 + d];
  SrS[r] = acc;
}

// ---- fills ------------------------------------------------------------------
__global__ void fill_f32(float* __restrict__ p, size_t n, float v) {
  size_t i = blockIdx.x * (size_t)blockDim.x + threadIdx.x;
  if (i < n) p[i] = v;
}
__global__ void fill_u32(unsigned* __restrict__ p, size_t n, unsigned v) {
  size_t i = blockIdx.x * (size_t)blockDim.x + threadIdx.x;
  if (i < n) p[i] = v;
}

// ---- edge phase 1: logits + segment max -------------------------------------
__global__ void edge_scores(const int* __restrict__ src, const int* __restrict__ tgt,
                            const int* __restrict__ rat,
                            const float* __restrict__ sT, const float* __restrict__ sS,
                            const float* __restrict__ sR, const float* __restrict__ a_b,
                            int E, float* __restrict__ ebuf, unsigned* __restrict__ m) {
  const int i = blockIdx.x * blockDim.x + threadIdx.x;
  if (i >= E) return;
  const int s = src[i], t = tgt[i];
  float e = sT[t] + sS[s] + a_b[0];
  if (rat) {
    int ri = rat[i] * 2 - 1;
    ri = ri < 0 ? 0 : (ri > 7 ? 7 : ri);
    e += sR[ri];
  }
  e = e > 0.0f ? e : 0.2f * e;   // leaky_relu(0.2)
  ebuf[i] = e;
  atomicMax(m + t, encOrd(e));
}

// ---- edge phase 2: exp + segment sum ----------------------------------------
__global__ void edge_exp(const int* __restrict__ tgt, const unsigned* __restrict__ m,
                         float* __restrict__ ebuf, float* __restrict__ den, int E) {
  const int i = blockIdx.x * blockDim.x + threadIdx.x;
  if (i >= E) return;
  const int t = tgt[i];
  const float ex = __expf(ebuf[i] - decOrd(m[t]));
  ebuf[i] = ex;
  atomicAdd(den + t, ex);
}

// ---- edge phase 3: weighted scatter of h_s ----------------------------------
__global__ void gat_scatter(const int* __restrict__ src, const int* __restrict__ tgt,
                            const int* __restrict__ rat, const float* __restrict__ Hsrc,
                            const float* __restrict__ HR, const float* __restrict__ ebuf,
                            const float* __restrict__ den, float* __restrict__ out, int E) {
  const int idx = blockIdx.x * blockDim.x + threadIdx.x;
  if (idx >= E * 16) return;
  const int e = idx >> 4;
  const int d4 = (idx & 15) << 2;
  const int t = tgt[e];
  const float alpha = ebuf[e] / (den[t] + 1e-16f);
  float4 v = *reinterpret_cast<const float4*>(Hsrc + (size_t)src[e] * DM + d4);
  if (rat) {
    int ri = rat[e] * 2 - 1;
    ri = ri < 0 ? 0 : (ri > 7 ? 7 : ri);
    const float4 hr = *reinterpret_cast<const float4*>(HR + ri * DM + d4);
    v.x += hr.x; v.y += hr.y; v.z += hr.z; v.w += hr.w;
  }
  float* o = out + (size_t)t * DM + d4;
  atomicAdd(o + 0, alpha * v.x);
  atomicAdd(o + 1, alpha * v.y);
  atomicAdd(o + 2, alpha * v.z);
  atomicAdd(o + 3, alpha * v.w);
}

// ---- batch feature build: x = [bu, bi, bu*bi] -------------------------------
__global__ void build_x(const int* __restrict__ uid, const int* __restrict__ iid,
                        const float* __restrict__ fu, const float* __restrict__ fi,
                        float* __restrict__ x, int B) {
  const int idx = blockIdx.x * blockDim.x + threadIdx.x;
  if (idx >= B * DM) return;
  const int b = idx >> 6, d = idx & 63;
  const float bu = fu[(size_t)uid[b] * DM + d];
  const float bi = fi[(size_t)iid[b] * DM + d];
  float* xr = x + (size_t)b * 192;
  xr[d] = bu;
  xr[64 + d] = bi;
  xr[128 + d] = bu * bi;
}

// ---- final 32->1 layer: wave-per-sample dot ---------------------------------
__global__ void final_dot(const float* __restrict__ x3, const float* __restrict__ w,
                          const float* __restrict__ b, float* __restrict__ out, int B) {
  const int gid = blockIdx.x * blockDim.x + threadIdx.x;
  const int wid = gid >> 5, lane = gid & 31;
  if (wid >= B) return;
  float v = x3[(size_t)wid * 32 + lane] * w[lane];
  for (int off = 16; off; off >>= 1) v += __shfl_down(v, off);
  if (lane == 0) out[wid] = v + b[0];
}

// ---------------------------------------------------------------------------
extern "C" void kernel_launch(void* const* d_in, const int* in_sizes, int n_in,
                              void* d_out, int out_size, void* d_ws, size_t ws_size,
                              hipStream_t stream) {
  // Inputs: setup_inputs order; params dict flattened in JAX pytree (sorted-key) order.
  const int* user_ids   = (const int*)d_in[0];
  const int* item_ids   = (const int*)d_in[1];
  const int* social_adj = (const int*)d_in[2];   // [2, NS]
  const int* inter_adj  = (const int*)d_in[3];   // [2, NE] (row0=user, row1=item)
  const int* inter_rat  = (const int*)d_in[4];
  const float* if_W  = (const float*)d_in[5];
  const float* if_b  = (const float*)d_in[6];
  const float* ii_W  = (const float*)d_in[7];
  const float* ii_ab = (const float*)d_in[8];
  const float* ii_aw = (const float*)d_in[9];
  const float* ii_b  = (const float*)d_in[10];
  const float* item_emb = (const float*)d_in[11];
  const float* p1_W = (const float*)d_in[12];
  const float* p1_b = (const float*)d_in[13];
  const float* p2_W = (const float*)d_in[14];
  const float* p2_b = (const float*)d_in[15];
  const float* p3_W = (const float*)d_in[16];
  const float* p3_b = (const float*)d_in[17];
  const float* p4_W = (const float*)d_in[18];
  const float* p4_b = (const float*)d_in[19];
  const float* rating_emb = (const float*)d_in[20];
  const float* soc_W  = (const float*)d_in[21];
  const float* soc_ab = (const float*)d_in[22];
  const float* soc_aw = (const float*)d_in[23];
  const float* soc_b  = (const float*)d_in[24];
  const float* uf_W = (const float*)d_in[25];
  const float* uf_b = (const float*)d_in[26];
  const float* ui_W  = (const float*)d_in[27];
  const float* ui_ab = (const float*)d_in[28];
  const float* ui_aw = (const float*)d_in[29];
  const float* ui_b  = (const float*)d_in[30];
  const float* user_emb = (const float*)d_in[31];

  const int B  = in_sizes[0];
  const int NS = in_sizes[2] / 2;
  const int NE = in_sizes[3] / 2;
  const int NI = in_sizes[11] / DM;
  const int NU = in_sizes[31] / DM;
  (void)n_in; (void)out_size; (void)ws_size;

  // --- workspace partition (256B aligned slabs) ---
  char* wsp = (char*)d_ws;
  auto alloc = [&](size_t nfloats) -> float* {
    float* p = (float*)wsp;
    wsp += ((nfloats * sizeof(float) + 255) & ~(size_t)255);
    return p;
  };
  float* bufA = alloc((size_t)NU * DM);   // H scratch (users) / fu
  float* bufB = alloc((size_t)NU * DM);   // user_social
  float* bufC = alloc((size_t)NI * DM);   // H scratch (items) / fi
  float* bufD = alloc((size_t)NU * DM);   // user_hist
  float* bufE = alloc((size_t)NI * DM);   // item_hist
  float* scoreT = alloc(NU);
  float* scoreS = alloc(NU);
  float* scoreJ = alloc(NU);              // dummy sink
  float* den    = alloc(NU);
  unsigned* mbuf = (unsigned*)alloc(NU);
  const int Emax = NS > NE ? NS : NE;
  float* ebuf  = alloc(Emax);
  float* HRui  = alloc(8 * DM);
  float* HRii  = alloc(8 * DM);
  float* SrSui = alloc(8);
  float* SrSii = alloc(8);
  float* xb = alloc((size_t)B * 192);
  float* x1 = alloc((size_t)B * 128);
  float* x2 = alloc((size_t)B * 64);
  float* x3 = alloc((size_t)B * 32);

  auto cdiv = [](long a, long b) -> int { return (int)((a + b - 1) / b); };

  auto run_gat = [&](const float* Hsrc, const float* HR, const float* SrS,
                     const int* src, const int* tgt, const int* rat,
                     const float* a_b, int E, int Ntgt, float* outbuf) {
    fill_f32<<<cdiv((long)Ntgt * DM, 256), 256, 0, stream>>>(outbuf, (size_t)Ntgt * DM, 0.0f);
    fill_f32<<<cdiv(Ntgt, 256), 256, 0, stream>>>(den, (size_t)Ntgt, 0.0f);
    fill_u32<<<cdiv(Ntgt, 256), 256, 0, stream>>>(mbuf, (size_t)Ntgt, 0x007FFFFFu); // enc(-inf)
    edge_scores<<<cdiv(E, 256), 256, 0, stream>>>(src, tgt, rat, scoreT, scoreS,
                                                  SrS, a_b, E, ebuf, mbuf);
    edge_exp<<<cdiv(E, 256), 256, 0, stream>>>(tgt, mbuf, ebuf, den, E);
    gat_scatter<<<cdiv((long)E * 16, 256), 256, 0, stream>>>(src, tgt, rat, Hsrc, HR,
                                                             ebuf, den, outbuf, E);
  };

  // --- rating transforms (shared by ui/ii GATs) ---
  rating_hr<<<2, 256, 0, stream>>>(rating_emb, ui_W, HRui);
  rating_hr<<<2, 256, 0, stream>>>(rating_emb, ii_W, HRii);
  rating_score<<<1, 8, 0, stream>>>(HRui, ui_aw, SrSui);
  rating_score<<<1, 8, 0, stream>>>(HRii, ii_aw, SrSii);

  // --- social GAT (users -> users) ---
  launch_gemm<DM, 0, 0>(user_emb, soc_W, soc_b, bufA, NU, DM, stream);
  node_scores<<<cdiv(NU, 8), 256, 0, stream>>>(bufA, soc_aw, NU, scoreT, scoreS);
  run_gat(bufA, nullptr, nullptr, social_adj, social_adj + NS, nullptr,
          soc_ab, NS, NU, bufB);

  // --- ui GAT (items -> users, with ratings) ---
  launch_gemm<DM, 0, 0>(user_emb, ui_W, ui_b, bufA, NU, DM, stream);   // H_t (users)
  node_scores<<<cdiv(NU, 8), 256, 0, stream>>>(bufA, ui_aw, NU, scoreT, scoreJ);
  launch_gemm<DM, 0, 0>(item_emb, ui_W, ui_b, bufC, NI, DM, stream);   // H_s (items)
  node_scores<<<cdiv(NI, 8), 256, 0, stream>>>(bufC, ui_aw, NI, scoreJ, scoreS);
  run_gat(bufC, HRui, SrSui, inter_adj + NE, inter_adj, inter_rat,
          ui_ab, NE, NU, bufD);

  // --- ii GAT (users -> items, with ratings) ---
  launch_gemm<DM, 0, 0>(item_emb, ii_W, ii_b, bufC, NI, DM, stream);   // H_t (items)
  node_scores<<<cdiv(NI, 8), 256, 0, stream>>>(bufC, ii_aw, NI, scoreT, scoreJ);
  launch_gemm<DM, 0, 0>(user_emb, ii_W, ii_b, bufA, NU, DM, stream);   // H_s (users)
  node_scores<<<cdiv(NU, 8), 256, 0, stream>>>(bufA, ii_aw, NU, scoreJ, scoreS);
  run_gat(bufA, HRii, SrSii, inter_adj, inter_adj + NE, inter_rat,
          ii_ab, NE, NI, bufE);

  // --- fusion: fu = relu([user_social, user_hist] @ uf_W + uf_b) -> bufA ---
  // concat GEMM as two passes: A0 @ W[0:64] + bias, then += A1 @ W[64:128], relu
  launch_gemm<DM, 0, 0>(bufB, uf_W,           uf_b,    bufA, NU, DM, stream);
  launch_gemm<DM, 1, 1>(bufD, uf_W + DM * DM, nullptr, bufA, NU, DM, stream);
  // --- fusion: fi = relu([item_emb, item_hist] @ if_W + if_b) -> bufC ---
  launch_gemm<DM, 0, 0>(item_emb, if_W,           if_b,    bufC, NI, DM, stream);
  launch_gemm<DM, 1, 1>(bufE,     if_W + DM * DM, nullptr, bufC, NI, DM, stream);

  // --- prediction MLP ---
  build_x<<<cdiv((long)B * DM, 256), 256, 0, stream>>>(user_ids, item_ids, bufA, bufC, xb, B);
  launch_gemm<192, 1, 0>(xb, p1_W, p1_b, x1, B, 128, stream);
  launch_gemm<128, 1, 0>(x1, p2_W, p2_b, x2, B, 64, stream);
  launch_gemm<DM, 1, 0>(x2, p3_W, p3_b, x3, B, 32, stream);
  final_dot<<<cdiv((long)B * 32, 256), 256, 0, stream>>>(x3, p4_W, p4_b, (float*)d_out, B);
}